// Ocean_e2e_37950331028269
// MI455X (gfx1250) — compile-verified
//
#include <hip/hip_runtime.h>

// ---------------------------------------------------------------------------
// Ocean advection + 3x3 binomial filter, 20 sequential steps.
// Memory-bound (~34 MB/step, fully L2-resident on MI455X's 192 MB L2), so:
//   * one fused kernel per step (advection + separable filter through LDS)
//   * 20 launches ping-pong d_ws <-> d_out (step 19 writes d_out)
//   * f32 throughout (matches reference numerics over 20 dependent steps)
// CDNA5 path: horizontal filter pass done as V_WMMA_F32_16X16X4_F32:
//   Out(16x16) = Sv(16x18, zero-padded to K=20) x Kx(20x16 banded const)
// ---------------------------------------------------------------------------

typedef __attribute__((ext_vector_type(2))) float v2f;
typedef __attribute__((ext_vector_type(8))) float v8f;

#define DT_STEP   600.0f
#define N_STEPS   20
#define R_EARTH   6371000.0f
#define DEG2RAD   0.017453292519943295f

#define TY 32            // output rows per block
#define TX 64            // output cols per block (8 waves x one 16x16 tile)
#define ST_H (TY + 4)    // Tc tile + halo 2            : 36
#define ST_W (TX + 6)    // 70 (feeds sTn cols 0..67)
#define SN_H (TY + 2)    // Tn tile + halo 1            : 34
#define SN_W (TX + 4)    // 68 (2 extra cols keep WMMA K-pad reads in bounds; Kx pad rows are 0)
#define SV_H TY          // vertically-filtered rows    : 32
#define SV_W (TX + 4)    // 68

#if defined(__has_builtin)
#if __has_builtin(__builtin_amdgcn_wmma_f32_16x16x4_f32)
#define HAVE_WMMA_F32X4 1
#endif
#endif

// Kx[j'][n] = w(j'-n), band weights {0.25, 0.5, 0.25}; rows j' >= 18 are pad.
__device__ __forceinline__ float kx_weight(int jp, int n) {
  if (jp >= 18) return 0.0f;
  int d = jp - n;
  if (d == 1) return 0.5f;
  if (d == 0 || d == 2) return 0.25f;
  return 0.0f;
}

__global__ __launch_bounds__(256) void ocean_step_kernel(
    const float* __restrict__ Tsrc, const float* __restrict__ ug,
    const float* __restrict__ vg, const float* __restrict__ lat,
    const float* __restrict__ lon, const float* __restrict__ mask,
    float* __restrict__ Tdst, int B, int H, int W)
{
  __shared__ float sT[ST_H][ST_W];    // Tc with halo 2 (x wrapped, y zeroed)
  __shared__ float sTn[SN_H][SN_W];   // advected Tn with halo 1 (zero outside)
  __shared__ float sV[SV_H][SV_W];    // vertical [1,2,1]/4 pass

  const int b   = blockIdx.z;
  const int ty0 = blockIdx.y * TY;
  const int tx0 = blockIdx.x * TX;
  const int tid = threadIdx.x;

  const float dlat        = lat[1] - lat[0];
  const float dlon        = lon[1] - lon[0];
  const float inv_dy      = 1.0f / (R_EARTH * DEG2RAD * fabsf(dlat));
  const float sign_y      = (dlat > 0.0f) ? 1.0f : -1.0f;
  const float inv2dx_base = 1.0f / (2.0f * R_EARTH * DEG2RAD * dlon);

  const size_t plane = (size_t)H * W;
  const float* Tb  = Tsrc + (size_t)b * plane;
  const float* ugb = ug   + (size_t)b * plane;
  const float* vgb = vg   + (size_t)b * plane;

  // ---- 1) stage Tc tile + halo 2 into LDS (x periodic, y zero-fill) ----
  for (int idx = tid; idx < ST_H * ST_W; idx += 256) {
    int i  = idx / ST_W, j = idx % ST_W;
    int gh = ty0 - 2 + i;
    int gw = tx0 - 2 + j;
    gw = ((gw % W) + W) % W;                      // periodic in x
    float v = 0.0f;
    if (gh >= 0 && gh < H) v = Tb[(size_t)gh * W + gw];
    sT[i][j] = v;
  }
  __syncthreads();

  // ---- 2) advection update -> Tn with halo 1 (zero outside domain) ----
  for (int idx = tid; idx < SN_H * SN_W; idx += 256) {
    int i  = idx / SN_W, j = idx % SN_W;
    int gh = ty0 - 1 + i;
    int gw = tx0 - 1 + j;
    float tn = 0.0f;                               // conv zero padding
    if (gh >= 0 && gh < H && gw >= 0 && gw < W) {
      float c  = sT[i + 1][j + 1];
      float xm = sT[i + 1][j];
      float xp = sT[i + 1][j + 2];
      float dTdy;
      if (gh == 0)          dTdy = (sT[i + 2][j + 1] - c) * inv_dy;
      else if (gh == H - 1) dTdy = (c - sT[i][j + 1]) * inv_dy;
      else                  dTdy = (sT[i + 2][j + 1] - sT[i][j + 1]) * (0.5f * inv_dy);
      dTdy *= sign_y;
      float clat = __cosf(lat[gh] * DEG2RAD);
      float dTdx = (xp - xm) * (inv2dx_base / clat);
      size_t g   = (size_t)gh * W + gw;
      float adv  = ugb[g] * dTdx + vgb[g] * dTdy;
      tn = c - DT_STEP * adv * mask[g];
    }
    sTn[i][j] = tn;
  }
  __syncthreads();

  // ---- 3) vertical binomial pass [1,2,1]/4 (VALU, exact f32) ----
  for (int idx = tid; idx < SV_H * SV_W; idx += 256) {
    int i = idx / SV_W, j = idx % SV_W;
    sV[i][j] = 0.25f * sTn[i][j] + 0.5f * sTn[i + 1][j] + 0.25f * sTn[i + 2][j];
  }
  __syncthreads();

  // ---- 4) horizontal pass: Out(16x16) = Sv(16x20 pad) x Kx(20x16) ----
  const int wave = tid >> 5;
  const int lane = tid & 31;
  const int r0   = (wave >> 2) * 16;   // wave's 16x16 tile inside block tile
  const int c0   = (wave & 3) * 16;
  const int m    = lane & 15;          // row (A) / column (B,D) within tile
  const int kk   = (lane >> 4) << 1;   // lanes 0-15 hold K0/K1, 16-31 K2/K3

#ifdef HAVE_WMMA_F32X4
  v8f acc = {};
  #pragma unroll
  for (int kb = 0; kb < 20; kb += 4) {
    v2f a, bm;
    // A: 16x4 f32 chunk of Sv (documented layout)
    a.x = sV[r0 + m][c0 + kb + kk];
    a.y = sV[r0 + m][c0 + kb + kk + 1];
    // B: 4x16 chunk of constant banded Kx (row-striped, VGPR0 = K0/K2 halves)
    bm.x = kx_weight(kb + kk,     m);
    bm.y = kx_weight(kb + kk + 1, m);
    acc = __builtin_amdgcn_wmma_f32_16x16x4_f32(
        /*neg_a=*/false, a, /*neg_b=*/false, bm,
        /*c_mod=*/(short)0, acc, /*reuse_a=*/false, /*reuse_b=*/false);
  }
  // D layout: VGPR v -> row v (lanes 0-15) / row v+8 (lanes 16-31), col = lane&15
  #pragma unroll
  for (int v = 0; v < 8; ++v) {
    int gh = ty0 + r0 + v + ((lane < 16) ? 0 : 8);
    int gw = tx0 + c0 + m;
    if (gh < H && gw < W) {
      size_t g = (size_t)gh * W + gw;
      Tdst[(size_t)b * plane + g] = acc[v] * mask[g];
    }
  }
#else
  // VALU fallback (same math) if the f32 WMMA builtin is unavailable
  for (int idx = tid; idx < TY * TX; idx += 256) {
    int i = idx / TX, j = idx % TX;
    int gh = ty0 + i, gw = tx0 + j;
    if (gh < H && gw < W) {
      float r = 0.25f * sV[i][j] + 0.5f * sV[i][j + 1] + 0.25f * sV[i][j + 2];
      size_t g = (size_t)gh * W + gw;
      Tdst[(size_t)b * plane + g] = r * mask[g];
    }
  }
#endif
}

extern "C" void kernel_launch(void* const* d_in, const int* in_sizes, int n_in,
                              void* d_out, int out_size, void* d_ws, size_t ws_size,
                              hipStream_t stream) {
  const float* T    = (const float*)d_in[0];
  const float* ug   = (const float*)d_in[1];
  const float* vg   = (const float*)d_in[2];
  const float* lat  = (const float*)d_in[3];
  const float* lon  = (const float*)d_in[4];
  const float* mask = (const float*)d_in[5];
  const int H = in_sizes[3];
  const int W = in_sizes[4];
  const int B = in_sizes[0] / (H * W);

  float* bufA = (float*)d_ws;    // even-step destination (needs B*H*W floats)
  float* out  = (float*)d_out;   // odd-step destination; step 19 (odd) = final

  dim3 grid((W + TX - 1) / TX, (H + TY - 1) / TY, B);
  dim3 block(256);

  const float* src = T;
  for (int s = 0; s < N_STEPS; ++s) {
    float* dst = (s & 1) ? out : bufA;
    hipLaunchKernelGGL(ocean_step_kernel, grid, block, 0, stream,
                       src, ug, vg, lat, lon, mask, dst, B, H, W);
    src = dst;
  }
}